// Pooling_38706245271888
// MI455X (gfx1250) — compile-verified
//
#include <hip/hip_runtime.h>

typedef __attribute__((ext_vector_type(2))) float v2f;
typedef __attribute__((ext_vector_type(4))) float v4f;
typedef __attribute__((ext_vector_type(8))) float v8f;
typedef int gv4i __attribute__((vector_size(16)));  // matches builtin param type

#define AS1 __attribute__((address_space(1)))
#define AS3 __attribute__((address_space(3)))

#if __has_builtin(__builtin_amdgcn_global_load_async_to_lds_b128)
#define HAVE_ASYNC_LDS 1
#endif

namespace {
constexpr int kB = 16, kT = 4096, kD = 768, kS = 64;
constexpr int kChunk = 16;
// 776 floats/row: 2*776 % 64 == 16 -> the two 16-lane half-wave fragment reads
// land on disjoint LDS banks.
constexpr int kRowStride = kD + 8;
}

// ---------------- rep branch: gather [CLS]-like rows ----------------
__global__ __launch_bounds__(256) void k_rep(const float* __restrict__ wv,
                                             const int* __restrict__ ids,
                                             const int* __restrict__ rmask,
                                             float* __restrict__ out) {
  int idx = blockIdx.x * blockDim.x + threadIdx.x;
  const int DV = kD / 4;
  if (idx >= kB * kS * DV) return;
  int d4 = idx % DV;
  int bs = idx / DV;
  int s = bs % kS, b = bs / kS;
  int tok = ids[b * kS + s];
  float m = (rmask[b * kS + s] != 0) ? 1.f : 0.f;
  const v4f* src = (const v4f*)(wv + ((size_t)b * kT + tok) * kD) + d4;
  v4f v = *src;
  v4f* dst = (v4f*)(out + ((size_t)b * 2 * kS + s) * kD) + d4;
  *dst = v * m;
}

// ---------------- output mask: [rep_mask | len_mask] as float ----------------
__global__ void k_mask(const int* __restrict__ rmask, const int* __restrict__ lmask,
                       float* __restrict__ out) {
  int i = blockIdx.x * blockDim.x + threadIdx.x;
  if (i >= kB * 2 * kS) return;
  int b = i / (2 * kS), j = i % (2 * kS);
  int v = (j < kS) ? rmask[b * kS + j] : lmask[b * kS + (j - kS)];
  out[(size_t)kB * 2 * kS * kD + i] = (v != 0) ? 1.f : 0.f;
}

// ---------------- mean branch: WMMA ones-matmul segment reduction ----------------
__global__ __launch_bounds__(256) void k_mean(const float* __restrict__ wv,
                                              const int* __restrict__ lens,
                                              const int* __restrict__ lmask,
                                              float* __restrict__ out) {
  __shared__ __align__(16) float tile[kChunk * kRowStride];
  __shared__ int anyflag;

  const int b = blockIdx.x / kS;
  const int s = blockIdx.x % kS;
  const int tid = (int)threadIdx.x;
  const int lane = tid & 31;
  const int wave = tid >> 5;   // 8 waves, 6 dim-tiles each -> 768 dims

  if (tid == 0) anyflag = 0;

  // Sentence span from cumulative lengths (searchsorted(cum, t, "right")):
  // sentence s owns tokens [cum[s-1], min(cum[s], T)).
  const int* lrow_len = lens + b * kS;
  int start = 0;
  for (int i = 0; i < s; ++i) start += lrow_len[i];
  int end = start + lrow_len[s];
  if (start > kT) start = kT;
  if (end > kT) end = kT;
  int span = end - start;
  if (span < 0) span = 0;

  const v8f zero8 = {0.f, 0.f, 0.f, 0.f, 0.f, 0.f, 0.f, 0.f};
  const v4f zero4 = {0.f, 0.f, 0.f, 0.f};
  v8f accS[6], accC[6];
#pragma unroll
  for (int i = 0; i < 6; ++i) { accS[i] = zero8; accC[i] = zero8; }
  v2f ones; ones.x = 1.f; ones.y = 1.f;

  const int r = tid >> 4;    // token row within chunk (0..15)
  const int c16 = tid & 15;  // 16 threads cooperate on one 768-float row

  const int nchunks = (span + kChunk - 1) / kChunk;
  for (int c = 0; c < nchunks; ++c) {
    const int t0 = start + c * kChunk;
    int nval = span - c * kChunk;
    if (nval > kChunk) nval = kChunk;

    float* lrow = tile + r * kRowStride;
    if (r < nval) {
      const float* gsrc = wv + ((size_t)b * kT + (size_t)(t0 + r)) * kD;
#pragma unroll
      for (int j = 0; j < 12; ++j) {
        int col = (c16 + j * 16) * 4;
#ifdef HAVE_ASYNC_LDS
        __builtin_amdgcn_global_load_async_to_lds_b128(
            (AS1 gv4i*)const_cast<float*>(gsrc + col),
            (AS3 gv4i*)(lrow + col), 0, 0);
#else
        *(v4f*)(lrow + col) = *(const v4f*)(gsrc + col);
#endif
      }
    } else {
#pragma unroll
      for (int j = 0; j < 12; ++j) {
        int col = (c16 + j * 16) * 4;
        *(v4f*)(lrow + col) = zero4;  // padding tokens contribute 0 to sum & count
      }
    }
#ifdef HAVE_ASYNC_LDS
#if __has_builtin(__builtin_amdgcn_s_wait_asynccnt)
    __builtin_amdgcn_s_wait_asynccnt(0);
#else
    asm volatile("s_wait_asynccnt 0" ::: "memory");
#endif
#endif
    __syncthreads();

    // A-fragment per ISA layout (32-bit A 16x4): lanes 0-15 hold K=0,1;
    // lanes 16-31 hold K=2,3; M = lane % 16 (here: M indexes 16 dims).
    const int half = lane >> 4;
    const int lr = lane & 15;
#pragma unroll
    for (int dt = 0; dt < 6; ++dt) {
      const int mbase = (wave * 6 + dt) * 16;
      const float* fb = tile + (2 * half) * kRowStride + mbase + lr;
#pragma unroll
      for (int ks = 0; ks < 4; ++ks) {
        v2f a;
        a.x = fb[ks * 4 * kRowStride];
        a.y = fb[ks * 4 * kRowStride + kRowStride];
        v2f ind;
        ind.x = (a.x != 0.f) ? 1.f : 0.f;
        ind.y = (a.y != 0.f) ? 1.f : 0.f;
        // D = A(16 dims x 4 tokens) * ones(4x16) + C : every column of D holds
        // the running token-sum for its dim row. Exact f32, matches reference.
        accS[dt] = __builtin_amdgcn_wmma_f32_16x16x4_f32(
            false, a, false, ones, (short)0, accS[dt], false, false);
        accC[dt] = __builtin_amdgcn_wmma_f32_16x16x4_f32(
            false, ind, false, ones, (short)0, accC[dt], false, false);
      }
    }
    __syncthreads();
  }

  // C/D layout: VGPR r, lanes 0-15 -> M=r, lanes 16-31 -> M=8+r (N replicated).
  const bool writerLo = (lane < 8);
  const bool writerHi = (lane >= 16 && lane < 24);
  const bool writer = writerLo || writerHi;
  const int comp = writerLo ? lane : (lane - 16);
  const int moff = writerLo ? comp : (8 + comp);

  union VU { v8f v; float f[8]; };

  if (writer) {
    float csum = 0.f;
#pragma unroll
    for (int dt = 0; dt < 6; ++dt) { VU u; u.v = accC[dt]; csum += u.f[comp]; }
    if (csum > 0.f) atomicOr(&anyflag, 1);
  }
  __syncthreads();
  const bool empty = (anyflag == 0);
  const float maskf = (lmask[b * kS + s] != 0) ? 1.f : 0.f;

  if (writer) {
#pragma unroll
    for (int dt = 0; dt < 6; ++dt) {
      const int dim = (wave * 6 + dt) * 16 + moff;
      VU us, uc;
      us.v = accS[dt];
      uc.v = accC[dt];
      float sum = us.f[comp], cnt = uc.f[comp];
      float den = (cnt == 0.f) ? 1.f : cnt;   // torch: divide by nonzero count
      float val = sum / den;
      if (empty) val = wv[dim];               // fallback to word_vectors[0,0,:]
      out[((size_t)b * 2 * kS + kS + s) * kD + dim] = val * maskf;
    }
  }
}

extern "C" void kernel_launch(void* const* d_in, const int* in_sizes, int n_in,
                              void* d_out, int out_size, void* d_ws, size_t ws_size,
                              hipStream_t stream) {
  (void)in_sizes; (void)n_in; (void)out_size; (void)d_ws; (void)ws_size;
  const float* wv   = (const float*)d_in[0];
  const int* ids    = (const int*)d_in[1];
  const int* rmask  = (const int*)d_in[2];
  const int* lens   = (const int*)d_in[3];
  const int* lmask  = (const int*)d_in[4];
  float* out = (float*)d_out;

  const int repElems = kB * kS * (kD / 4);
  k_rep<<<(repElems + 255) / 256, 256, 0, stream>>>(wv, ids, rmask, out);
  k_mean<<<kB * kS, 256, 0, stream>>>(wv, lens, lmask, out);
  k_mask<<<(kB * 2 * kS + 255) / 256, 256, 0, stream>>>(rmask, lmask, out);
}